// RTDFRCNNBackBone_Concat_34059090657835
// MI455X (gfx1250) — compile-verified
//
#include <hip/hip_runtime.h>
#include <stdint.h>

// CDNA5 / gfx1250, wave32.
typedef float v2f __attribute__((ext_vector_type(2)));
typedef float v8f __attribute__((ext_vector_type(8)));

#define IMG_W   1248
#define IMG_H   376
#define IMG_HW  (IMG_W * IMG_H)
#define IMG_CHW (16 * IMG_HW)
#define WAVES_PER_BLOCK 8

// Each wave processes a tile of 16 voxels:
//  - projection of all 16 voxels vs all 4 calib matrices in ONE
//    v_wmma_f32_16x16x4_f32 (A = pts_h 16x4, B = calib columns 4x16)
//  - image-channel gather: lanes 0-15 serve voxel j, lanes 16-31 voxel j+8
//  - lidar feature copy staged through LDS with async global<->LDS DMA
__global__ __launch_bounds__(256) void fuse_voxel_image_kernel(
    const float* __restrict__ vf,      // (N,16) lidar features
    const int*   __restrict__ coords,  // (N,4)  b,z,y,x
    const float* __restrict__ images,  // (4,16,376,1248)
    const float* __restrict__ calib,   // (4,3,4)
    float*       __restrict__ out,     // (N,32)
    int nvox)
{
    __shared__ __align__(16) float lds_tile[WAVES_PER_BLOCK][256]; // 1KB/wave

    const int lane = threadIdx.x & 31;
    const int wid  = threadIdx.x >> 5;
    const int gw   = blockIdx.x * WAVES_PER_BLOCK + wid;
    const int nw   = gridDim.x * WAVES_PER_BLOCK;
    const int ntiles = nvox >> 4;      // N is a multiple of 16 (300000)

    const int n    = lane & 15;        // column / voxel-in-halftile index
    const int half = lane >> 4;        // 0: K=0,1 half  1: K=2,3 half

    // ---- B matrix (wave-invariant): B[k, col], col = 4*b + i (i<3), slot 3 = 0
    // Layout: VGPR0 lane n -> B[0,n], lane 16+n -> B[2,n]
    //         VGPR1 lane n -> B[1,n], lane 16+n -> B[3,n]
    const int kb = half * 2;
    const int bb = n >> 2, ii = n & 3;
    v2f Bm;
    Bm.x = (ii < 3) ? calib[bb * 12 + ii * 4 + kb]     : 0.0f;
    Bm.y = (ii < 3) ? calib[bb * 12 + ii * 4 + kb + 1] : 0.0f;

    // Low 32 bits of the generic address of a __shared__ object are the
    // group-segment byte offset -> usable as the async-op LDS address VGPR.
    const unsigned lds_base = (unsigned)(uintptr_t)(&lds_tile[wid][0]);
    const int c = n; // image channel gathered by this lane
    const float* img_c = images + (size_t)c * IMG_HW;  // hoist channel offset

    for (int tile = gw; tile < ntiles; tile += nw) {
        const int n0 = tile << 4;

        // ---- kick off async DMA of this tile's lidar features into LDS
        // (overlaps with WMMA + gather below; tracked by ASYNCcnt)
        {
            const float* gsrc = vf + (size_t)n0 * 16;
            #pragma unroll
            for (int it = 0; it < 2; ++it) {
                unsigned chunk = (unsigned)(lane + 32 * it);   // 0..63 x 16B
                unsigned loff  = lds_base + chunk * 16u;
                const float* ga = gsrc + chunk * 4u;
                asm volatile("global_load_async_to_lds_b128 %0, %1, off"
                             :: "v"(loff), "v"(ga) : "memory");
            }
        }

        // ---- A matrix: pts_h for voxel (n0+n)
        // pts_h = (X, Y, Z, 1) with X=x*0.05, Y=y*0.05-25.6, Z=z*0.1-2.0
        // Layout: lane m  -> A[m,0..1] in v2f, lane 16+m -> A[m,2..3]
        int4  cd   = ((const int4*)coords)[(size_t)(n0 + n)];
        int   bidx = cd.x;
        float X = (float)cd.w * 0.05f;
        float Y = (float)cd.z * 0.05f - 25.6f;
        float Z = (float)cd.y * 0.10f - 2.0f;
        v2f Am;
        Am.x = half ? Z    : X;
        Am.y = half ? 1.0f : Y;

        // ---- D[m, 4b+i] = proj component i of voxel m against calib b
        v8f acc = {};
        v8f d = __builtin_amdgcn_wmma_f32_16x16x4_f32(
                    false, Am, false, Bm, (short)0, acc, false, false);

        // ---- gather phase: step j handles voxels j (low half) and j+8 (high)
        #pragma unroll
        for (int j = 0; j < 8; ++j) {
            int   vox = j + 8 * half;                 // voxel served (0..15)
            int   vb  = __shfl(bidx, vox, 32);        // its batch index
            int   sb  = 16 * half + 4 * vb;           // D source lane base
            float dj  = d[j];                         // D row vox lives in VGPR j
            float pu  = __shfl(dj, sb + 0, 32);
            float pv  = __shfl(dj, sb + 1, 32);
            float pw  = __shfl(dj, sb + 2, 32);
            // fast perspective divide: pw = X+1 >= 1 here; v_rcp_f32 (~1ulp)
            // instead of the ~20-op IEEE div chain — shortens the dependency
            // path from ds_bpermute results to the gather-load issue.
            float rw  = __builtin_amdgcn_rcpf(pw);
            float fu  = pu * rw;
            float fv  = pv * rw;
            int u = (int)fu;
            int v = (int)fv;
            bool valid = (u >= 0) & (u < IMG_W) & (v >= 0) & (v < IMG_H);
            int uc = min(max(u, 0), IMG_W - 1);
            int vc = min(max(v, 0), IMG_H - 1);
            float val = 0.0f;
            if (valid)  // served from 192MB L2: whole image tensor (~120MB) fits
                val = img_c[(size_t)vb * IMG_CHW + (size_t)vc * IMG_W + uc];
            out[(size_t)(n0 + vox) * 32 + c] = val;   // coalesced 64B per half
        }

        // ---- async loads must land before async stores read the LDS tile
        asm volatile("s_wait_asynccnt 0x0" ::: "memory");

        // ---- DMA lidar half out: out[(n0+vox)*32 + 16 .. 31]
        #pragma unroll
        for (int it = 0; it < 2; ++it) {
            unsigned idx = (unsigned)(lane + 32 * it);  // 0..63 x 16B chunks
            unsigned vox = idx >> 2, k = idx & 3;
            unsigned loff = lds_base + idx * 16u;
            float* gd = out + (size_t)(n0 + vox) * 32 + 16 + k * 4;
            asm volatile("global_store_async_from_lds_b128 %0, %1, off"
                         :: "v"(gd), "v"(loff) : "memory");
        }
        // LDS buffer reused next iteration: wait for stores to drain
        asm volatile("s_wait_asynccnt 0x0" ::: "memory");
    }
}

extern "C" void kernel_launch(void* const* d_in, const int* in_sizes, int n_in,
                              void* d_out, int out_size, void* d_ws, size_t ws_size,
                              hipStream_t stream) {
    const float* vf     = (const float*)d_in[0];
    const int*   coords = (const int*)d_in[1];
    const float* images = (const float*)d_in[2];
    const float* calib  = (const float*)d_in[3];
    float*       out    = (float*)d_out;

    int nvox   = in_sizes[0] / 16;          // (N,16) lidar features
    int ntiles = nvox / 16;                 // 18750 for N=300000
    int blocks = (ntiles + WAVES_PER_BLOCK - 1) / WAVES_PER_BLOCK;
    if (blocks < 1) blocks = 1;

    fuse_voxel_image_kernel<<<blocks, 256, 0, stream>>>(
        vf, coords, images, calib, out, nvox);
}